// HiPPOScale_9646496547005
// MI455X (gfx1250) — compile-verified
//
#include <hip/hip_runtime.h>

// HiPPO-LegS scan, chunked parallel-scan formulation for MI455X (gfx1250).
//   Phase 1: per-chunk local carry + cumulative transition P^T (8 waves, wave-private rows)
//   Phase 2: sequential chunk combine (1 block)
//   Phase 3: per-chunk replay with correct incoming carry, stream outputs
// 64x64x64 fp32 GEMMs on V_WMMA_F32_16X16X4_F32; A_t staging via
// GLOBAL_LOAD_ASYNC_TO_LDS_B128 (ASYNCcnt) when the toolchain exposes it.

typedef float v2f __attribute__((ext_vector_type(2)));
typedef float v4f __attribute__((ext_vector_type(4)));
typedef float v8f __attribute__((ext_vector_type(8)));
typedef int   v4i __attribute__((ext_vector_type(4)));

#define NDIM   64
#define LDIM   16384
#define BDIM   64
#define CHUNKS 256
#define TSTEPS 64              // LDIM / CHUNKS
#define MAT    (NDIM * NDIM)   // 4096 floats = 16KB
#define BUFSZ  (MAT + 128)     // A_t + f row + Bst row

// Workspace layout (floats):
//   P      : [0,              CHUNKS*MAT)
//   clocal : [CHUNKS*MAT,   2*CHUNKS*MAT)
//   cstart : [2*CHUNKS*MAT, 3*CHUNKS*MAT)     total ~12.6 MB

// --------------------------- async global->LDS ------------------------------
#if defined(__has_builtin)
#if __has_builtin(__builtin_amdgcn_global_load_async_to_lds_b128) && \
    __has_builtin(__builtin_amdgcn_global_load_async_to_lds_b32)
#define USE_ASYNC 1
#endif
#endif
#ifndef USE_ASYNC
#define USE_ASYNC 0
#endif

#if USE_ASYNC
__device__ __forceinline__ void async_b128(const float* g, float* l) {
    __builtin_amdgcn_global_load_async_to_lds_b128(
        (__attribute__((address_space(1))) v4i*)g,
        (__attribute__((address_space(3))) v4i*)l, 0, 0);
}
__device__ __forceinline__ void async_b32(const float* g, float* l) {
    __builtin_amdgcn_global_load_async_to_lds_b32(
        (__attribute__((address_space(1))) int*)g,
        (__attribute__((address_space(3))) int*)l, 0, 0);
}
__device__ __forceinline__ void wait_async0() {
#if __has_builtin(__builtin_amdgcn_s_wait_asynccnt)
    __builtin_amdgcn_s_wait_asynccnt(0);
#else
    asm volatile("s_wait_asynccnt 0x0" ::: "memory");
#endif
}
// Stage A_t (4096 floats) + f_t (64) + Bst_t (64) into LDS buf via async DMA.
template <int NT>
__device__ __forceinline__ void stage_async(const float* __restrict__ A_t,
                                            const float* __restrict__ f_t,
                                            const float* __restrict__ bst_t,
                                            float* __restrict__ buf, int tid)
{
    constexpr int NV4 = MAT / 4 / NT;
#pragma unroll
    for (int i = 0; i < NV4; ++i)
        async_b128(A_t + (size_t)(tid + i * NT) * 4, buf + (size_t)(tid + i * NT) * 4);
    if (tid < 64)       async_b32(f_t + tid, buf + MAT + tid);
    else if (tid < 128) async_b32(bst_t + (tid - 64), buf + MAT + tid);
}
#endif

// ----------------------- register-pipeline fallback -------------------------
template <int NT>
__device__ __forceinline__ void stage_load(const float* __restrict__ A_t,
                                           const float* __restrict__ f_t,
                                           const float* __restrict__ bst_t,
                                           v4f* r, float* rf, int tid)
{
    constexpr int NV4 = MAT / 4 / NT;
#pragma unroll
    for (int i = 0; i < NV4; ++i)
        r[i] = *(const v4f*)(A_t + (size_t)(tid + i * NT) * 4);
    if (tid < 64)        *rf = f_t[tid];
    else if (tid < 128)  *rf = bst_t[tid - 64];
}

template <int NT>
__device__ __forceinline__ void stage_store(float* __restrict__ buf,
                                            const v4f* r, float rf, int tid)
{
    constexpr int NV4 = MAT / 4 / NT;
#pragma unroll
    for (int i = 0; i < NV4; ++i)
        *(v4f*)(buf + (size_t)(tid + i * NT) * 4) = r[i];
    if (tid < 128) buf[MAT + tid] = rf;
}

// ---------------------------------------------------------------------------
// One recurrence step for one wave owning 16 state rows:
//   state_rows = state_rows @ A_t^T (+ accInit).  Abuf row-major: B[k][n]=Abuf[n*64+k].
// ---------------------------------------------------------------------------
__device__ __forceinline__ void wmma_step(const float* __restrict__ Abuf,
                                          float* __restrict__ carry,
                                          int rowbase, int lane,
                                          const v8f* accInit, v8f* accOut)
{
    const int lh = lane & 15;
    const int hi = lane >> 4;

    v2f afrag[16];
#pragma unroll
    for (int ks = 0; ks < 16; ++ks) {
        const float* p = &carry[(rowbase + lh) * NDIM + 4 * ks + 2 * hi];
        afrag[ks] = *(const v2f*)p;
    }

#pragma unroll
    for (int nt = 0; nt < 4; ++nt) {
        v8f acc = accInit[nt];
#pragma unroll
        for (int ks = 0; ks < 16; ++ks) {
            const float* p = &Abuf[(nt * 16 + lh) * NDIM + 4 * ks + 2 * hi];
            v2f b = *(const v2f*)p;
            acc = __builtin_amdgcn_wmma_f32_16x16x4_f32(
                false, afrag[ks], false, b, (short)0, acc, false, false);
        }
        accOut[nt] = acc;
    }

#pragma unroll
    for (int nt = 0; nt < 4; ++nt)
#pragma unroll
        for (int v = 0; v < 8; ++v)
            carry[(rowbase + v + 8 * hi) * NDIM + nt * 16 + lh] = accOut[nt][v];
}

// ---------------------------------------------------------------------------
// Phase 1: per-chunk local scan.  Waves 0-3 evolve carry (u added),
// waves 4-7 evolve X = P^T (identity init).  State rows are wave-private.
// Dynamic LDS: 2*BUFSZ (double-buffered staging) + 128*64 state  (~65 KB).
// ---------------------------------------------------------------------------
__global__ __launch_bounds__(256) void hippo_phase1(const float* __restrict__ inputs,
                                                    const float* __restrict__ A_st,
                                                    const float* __restrict__ B_st,
                                                    float* __restrict__ ws)
{
    extern __shared__ float smem[];
    float* buf0  = smem;
    float* buf1  = smem + BUFSZ;
    float* carry = smem + 2 * BUFSZ;   // 128 x 64

    const int g    = blockIdx.x;
    const int tid  = threadIdx.x;
    const int lane = tid & 31;
    const int wave = tid >> 5;
    const int rowbase = wave * 16;
    const bool hasU = (wave < 4);
    const int lh = lane & 15, hi = lane >> 4;

    for (int i = tid; i < MAT; i += 256) carry[i] = 0.0f;
    for (int i = tid; i < MAT; i += 256) {
        int rr = i >> 6, cc = i & 63;
        carry[MAT + i] = (rr == cc) ? 1.0f : 0.0f;
    }

    const size_t t0 = (size_t)g * TSTEPS;
#if USE_ASYNC
    stage_async<256>(A_st + t0 * MAT, inputs + t0 * BDIM, B_st + t0 * NDIM, buf0, tid);
#else
    v4f r[4]; float rf = 0.0f;
    stage_load<256>(A_st + t0 * MAT, inputs + t0 * BDIM, B_st + t0 * NDIM, r, &rf, tid);
#endif

    v8f acc[4], init[4];
    for (int s = 0; s < TSTEPS; ++s) {
        float* cur = (s & 1) ? buf1 : buf0;
        float* nxt = (s & 1) ? buf0 : buf1;
#if USE_ASYNC
        wait_async0();                      // DMA for `cur` complete (own wave)
#else
        stage_store<256>(cur, r, rf, tid);
#endif
        __syncthreads();                    // all waves' staging visible
        if (s + 1 < TSTEPS) {               // overlap next-step DMA with WMMAs
            const size_t t = t0 + s + 1;
#if USE_ASYNC
            stage_async<256>(A_st + t * MAT, inputs + t * BDIM, B_st + t * NDIM, nxt, tid);
#else
            stage_load<256>(A_st + t * MAT, inputs + t * BDIM, B_st + t * NDIM, r, &rf, tid);
            (void)nxt;
#endif
        }
        if (hasU) {                         // wave-uniform branch, EXEC stays full
            float fv[8];
#pragma unroll
            for (int v = 0; v < 8; ++v) fv[v] = cur[MAT + rowbase + v + 8 * hi];
#pragma unroll
            for (int nt = 0; nt < 4; ++nt) {
                float bs = cur[MAT + 64 + nt * 16 + lh];
#pragma unroll
                for (int v = 0; v < 8; ++v) init[nt][v] = fv[v] * bs;
            }
        } else {
#pragma unroll
            for (int nt = 0; nt < 4; ++nt)
#pragma unroll
                for (int v = 0; v < 8; ++v) init[nt][v] = 0.0f;
        }
        wmma_step(cur, carry, rowbase, lane, init, acc);
    }
    __syncthreads();

    float* clocal = ws + (size_t)CHUNKS * MAT + (size_t)g * MAT;
    float* Pg     = ws + (size_t)g * MAT;
    for (int i = tid; i < MAT; i += 256) clocal[i] = carry[i];
    for (int i = tid; i < MAT; i += 256) {
        int rr = i >> 6, cc = i & 63;              // X[rr][cc] = P[cc][rr]
        Pg[(size_t)cc * NDIM + rr] = carry[MAT + i];
    }
}

// ---------------------------------------------------------------------------
// Phase 2: sequential combine: C_start[g+1] = C_start[g] @ P_g^T + C_local[g]
// ---------------------------------------------------------------------------
__global__ __launch_bounds__(128) void hippo_phase2(float* __restrict__ ws)
{
    __shared__ float Abuf[MAT];
    __shared__ float carry[MAT];

    const int tid  = threadIdx.x;
    const int lane = tid & 31;
    const int wave = tid >> 5;
    const int rowbase = wave * 16;
    const int lh = lane & 15, hi = lane >> 4;

    for (int i = tid; i < MAT; i += 128) carry[i] = 0.0f;
    __syncthreads();

    const float* Pbase  = ws;
    const float* clocal = ws + (size_t)CHUNKS * MAT;
    float*       cstart = ws + (size_t)2 * CHUNKS * MAT;

    v8f acc[4], init[4];
    for (int g = 0; g < CHUNKS; ++g) {
        for (int i = tid; i < MAT; i += 128) {
            cstart[(size_t)g * MAT + i] = carry[i];
            Abuf[i] = Pbase[(size_t)g * MAT + i];
        }
#pragma unroll
        for (int nt = 0; nt < 4; ++nt)
#pragma unroll
            for (int v = 0; v < 8; ++v)
                init[nt][v] = clocal[(size_t)g * MAT +
                                     (size_t)(rowbase + v + 8 * hi) * NDIM + nt * 16 + lh];
        __syncthreads();
        wmma_step(Abuf, carry, rowbase, lane, init, acc);
        __syncthreads();
    }
}

// ---------------------------------------------------------------------------
// Phase 3: replay each chunk with correct incoming carry, stream outputs.
// Static LDS: 2*BUFSZ + MAT = ~49 KB.
// ---------------------------------------------------------------------------
__global__ __launch_bounds__(128) void hippo_phase3(const float* __restrict__ inputs,
                                                    const float* __restrict__ A_st,
                                                    const float* __restrict__ B_st,
                                                    const float* __restrict__ ws,
                                                    float* __restrict__ out)
{
    __shared__ float buf[2 * BUFSZ];
    __shared__ float carry[MAT];

    const int g    = blockIdx.x;
    const int tid  = threadIdx.x;
    const int lane = tid & 31;
    const int wave = tid >> 5;
    const int rowbase = wave * 16;
    const int lh = lane & 15, hi = lane >> 4;

    const float* cstart = ws + (size_t)2 * CHUNKS * MAT + (size_t)g * MAT;
    for (int i = tid; i < MAT; i += 128) carry[i] = cstart[i];

    const size_t t0 = (size_t)g * TSTEPS;
#if USE_ASYNC
    stage_async<128>(A_st + t0 * MAT, inputs + t0 * BDIM, B_st + t0 * NDIM, buf, tid);
#else
    v4f r[8]; float rf = 0.0f;
    stage_load<128>(A_st + t0 * MAT, inputs + t0 * BDIM, B_st + t0 * NDIM, r, &rf, tid);
#endif

    v8f acc[4], init[4];
    for (int s = 0; s < TSTEPS; ++s) {
        float* cur = buf + (s & 1) * BUFSZ;
        float* nxt = buf + ((s + 1) & 1) * BUFSZ;
#if USE_ASYNC
        wait_async0();
#else
        stage_store<128>(cur, r, rf, tid);
#endif
        __syncthreads();
        if (s + 1 < TSTEPS) {
            const size_t t = t0 + s + 1;
#if USE_ASYNC
            stage_async<128>(A_st + t * MAT, inputs + t * BDIM, B_st + t * NDIM, nxt, tid);
#else
            stage_load<128>(A_st + t * MAT, inputs + t * BDIM, B_st + t * NDIM, r, &rf, tid);
            (void)nxt;
#endif
        }
        float fv[8];
#pragma unroll
        for (int v = 0; v < 8; ++v) fv[v] = cur[MAT + rowbase + v + 8 * hi];
#pragma unroll
        for (int nt = 0; nt < 4; ++nt) {
            float bs = cur[MAT + 64 + nt * 16 + lh];
#pragma unroll
            for (int v = 0; v < 8; ++v) init[nt][v] = fv[v] * bs;
        }
        wmma_step(cur, carry, rowbase, lane, init, acc);

        float* o = out + (t0 + s) * (size_t)(BDIM * NDIM);
#pragma unroll
        for (int nt = 0; nt < 4; ++nt)
#pragma unroll
            for (int v = 0; v < 8; ++v)
                o[(size_t)(rowbase + v + 8 * hi) * NDIM + nt * 16 + lh] = acc[nt][v];
    }
}

// ---------------------------------------------------------------------------
extern "C" void kernel_launch(void* const* d_in, const int* in_sizes, int n_in,
                              void* d_out, int out_size, void* d_ws, size_t ws_size,
                              hipStream_t stream)
{
    const float* inputs = (const float*)d_in[0];   // (L, B)
    const float* A_st   = (const float*)d_in[1];   // (L, N, N)
    const float* B_st   = (const float*)d_in[2];   // (L, N)
    float*       out    = (float*)d_out;           // (L, B, N)
    float*       ws     = (float*)d_ws;            // needs >= 12.6 MB

    const size_t p1_lds = (size_t)(2 * BUFSZ + 2 * MAT) * sizeof(float);  // ~65 KB
    hippo_phase1<<<CHUNKS, 256, p1_lds, stream>>>(inputs, A_st, B_st, ws);
    hippo_phase2<<<1, 128, 0, stream>>>(ws);
    hippo_phase3<<<CHUNKS, 128, 0, stream>>>(inputs, A_st, B_st, ws, out);
}